// KroneckerLinear_48395691491923
// MI455X (gfx1250) — compile-verified
//
#include <hip/hip_runtime.h>

typedef float v2f __attribute__((ext_vector_type(2)));
typedef float v8f __attribute__((ext_vector_type(8)));
typedef int   vi4 __attribute__((ext_vector_type(4)));

#define AS1 __attribute__((address_space(1)))
#define AS3 __attribute__((address_space(3)))

#define LDS_PAD 68            // row stride (dwords): 68 % 64 == 4 -> conflict-free b64 operand gathers
#define A_OFF   0
#define B_OFF   (64 * LDS_PAD)
#define X0_OFF  (2 * 64 * LDS_PAD)
#define XBUF_DW (64 * LDS_PAD)
#define SMEM_DW (4 * 64 * LDS_PAD)   // 17408 dwords = 69632 bytes (A, B, X double-buffer)

__device__ __forceinline__ float swapx16(float v) {
    // ds_swizzle group-of-32: and=0x1f, or=0, xor=0x10  -> lanes L <-> L^16
    return __int_as_float(__builtin_amdgcn_ds_swizzle(__float_as_int(v), 0x401F));
}

// async global -> LDS, 16 bytes per lane (ASYNCcnt-tracked)
__device__ __forceinline__ void async_b128(const float* g, float* l) {
#if __has_builtin(__builtin_amdgcn_global_load_async_to_lds_b128)
    __builtin_amdgcn_global_load_async_to_lds_b128(
        (AS1 vi4*)g, (AS3 vi4*)l, 0, 0);
#else
    unsigned loff = (unsigned)(size_t)(AS3 void*)l;
    asm volatile("global_load_async_to_lds_b128 %0, %1, off"
                 :: "v"(loff), "v"(g) : "memory");
#endif
}

__device__ __forceinline__ void wait_async0() {
#if __has_builtin(__builtin_amdgcn_s_wait_asynccnt)
    __builtin_amdgcn_s_wait_asynccnt(0);
#else
    asm volatile("s_wait_asynccnt 0" ::: "memory");
#endif
}

__global__ __launch_bounds__(128) void kron_f32_wmma_kernel(
    const float* __restrict__ x,    // [M, 4096]
    const float* __restrict__ Aw,   // [64, 64]
    const float* __restrict__ Bw,   // [64, 64]
    const float* __restrict__ bias, // [4096]
    float* __restrict__ out,        // [M, 4096]
    int m_per_wg, int M)
{
    __shared__ float smem[SMEM_DW];

    const int tid  = threadIdx.x;
    const int lane = tid & 31;          // wave32
    const int wave = tid >> 5;          // wave owns q-tile qt == wave
    const int L    = lane & 15;
    const int h    = (lane >> 4) & 1;
    const bool low = (lane < 16);

    // ---- stage A/B weights into padded LDS (cooperative, once) ----
    #pragma unroll
    for (int i = 0; i < 8; ++i) {
        int idx = i * 128 + tid;            // float4 chunk id, 0..1023
        int k   = idx >> 4;                 // row 0..63
        int l   = (idx & 15) * 4;           // col 0..60
        float4 av = ((const float4*)Aw)[idx];
        float4 bv = ((const float4*)Bw)[idx];
        *(float4*)&smem[A_OFF + k * LDS_PAD + l] = av;
        *(float4*)&smem[B_OFF + k * LDS_PAD + l] = bv;
    }

    // ---- bias slice for this wave lives in registers: breg[pt][r] ----
    float breg[4][8];
    #pragma unroll
    for (int pt = 0; pt < 4; ++pt)
        #pragma unroll
        for (int r = 0; r < 8; ++r)
            breg[pt][r] = bias[(pt * 16 + r + 8 * h) * 64 + wave * 16 + L];

    const int m0 = blockIdx.x * m_per_wg;

    __syncthreads();

    // ---- prologue: async-stage first x row into buffer 0 ----
    {
        const float* xr = x + (size_t)m0 * 4096;
        #pragma unroll
        for (int i = 0; i < 8; ++i) {
            int idx = i * 128 + tid;
            int k   = idx >> 4;
            int l   = (idx & 15) * 4;
            async_b128(xr + idx * 4, &smem[X0_OFF + k * LDS_PAD + l]);
        }
    }

    for (int mi = 0; mi < m_per_wg; ++mi) {
        const int m = m0 + mi;
        if (m >= M) break;                       // block-uniform
        const int xo = X0_OFF + (mi & 1) * XBUF_DW;

        wait_async0();        // my async copies into current buffer are done
        __syncthreads();      // everyone's copies landed; prior readers of other buffer done

        // ---- kick off async staging of the NEXT x row into the other buffer ----
        if (mi + 1 < m_per_wg && m + 1 < M) {
            const float* xn = x + (size_t)(m + 1) * 4096;
            const int xno = X0_OFF + ((mi + 1) & 1) * XBUF_DW;
            #pragma unroll
            for (int i = 0; i < 8; ++i) {
                int idx = i * 128 + tid;
                int k   = idx >> 4;
                int l   = (idx & 15) * 4;
                async_b128(xn + idx * 4, &smem[xno + k * LDS_PAD + l]);
            }
        }

        // ---- stage A:  U[k][q] = sum_l Xm[k][l] * B[q][l],  wave owns q-strip ----
        // u[kt] : D-layout tile, lane(16h+L) vgpr r -> U[16kt + r + 8h][16*wave + L]
        v8f u[4] = {};
        const int brow = (wave * 16 + L) * LDS_PAD;
        #pragma unroll
        for (int ls = 0; ls < 16; ++ls) {
            const int col = ls * 4 + h * 2;
            v2f bop = *(const v2f*)&smem[B_OFF + brow + col];   // B[16w+L][4ls+2h+j]
            #pragma unroll
            for (int kt = 0; kt < 4; ++kt) {
                v2f aop = *(const v2f*)&smem[xo + (kt * 16 + L) * LDS_PAD + col];
                u[kt] = __builtin_amdgcn_wmma_f32_16x16x4_f32(
                            false, aop, false, bop, (short)0, u[kt], false, false);
            }
        }

        // ---- in-register relayout: D-layout -> B-operand pieces (SWAPX16 + half-select)
        // w_[kt][idx(kl,j)] holds, at lane 16h+L:  U[16kt + 4kl + 2h + j][16w+L]
        // idx(kl,j) = ((kl&1)<<2) | (kl&2) | j
        float w_[4][8];
        #pragma unroll
        for (int kt = 0; kt < 4; ++kt) {
            #pragma unroll
            for (int base = 0; base < 8; base += 4) {
                #pragma unroll
                for (int j = 0; j < 2; ++j) {
                    float a  = u[kt][base + j];
                    float b  = u[kt][base + j + 2];
                    float sa = swapx16(a);
                    float sb = swapx16(b);
                    w_[kt][base + j]     = low ? a  : sb;   // op(kl = base?1:0, j)
                    w_[kt][base + j + 2] = low ? sa : b;    // op(kl = base?3:2, j)
                }
            }
        }

        // ---- stage B:  Out[p][16w+L] = sum_k A[p][k] * U[k][16w+L]  ----
        const size_t obase = (size_t)m * 4096 + wave * 16 + L;
        #pragma unroll
        for (int pt = 0; pt < 4; ++pt) {
            v8f acc = {};
            const int arow = (pt * 16 + L) * LDS_PAD;
            #pragma unroll
            for (int ks = 0; ks < 16; ++ks) {
                const int kt = ks >> 2, kl = ks & 3;
                v2f aop = *(const v2f*)&smem[A_OFF + arow + ks * 4 + h * 2];
                const int i0 = ((kl & 1) << 2) | (kl & 2);
                v2f bop;
                bop.x = w_[kt][i0];
                bop.y = w_[kt][i0 + 1];
                acc = __builtin_amdgcn_wmma_f32_16x16x4_f32(
                          false, aop, false, bop, (short)0, acc, false, false);
            }
            #pragma unroll
            for (int r = 0; r < 8; ++r)
                out[obase + (size_t)(pt * 16 + r + 8 * h) * 64] = acc[r] + breg[pt][r];
        }
        // no trailing barrier needed: next iteration starts with wait_async + barrier
    }
}

extern "C" void kernel_launch(void* const* d_in, const int* in_sizes, int n_in,
                              void* d_out, int out_size, void* d_ws, size_t ws_size,
                              hipStream_t stream) {
    const float* x    = (const float*)d_in[0];
    const float* A    = (const float*)d_in[1];
    const float* B    = (const float*)d_in[2];
    const float* bias = (const float*)d_in[3];
    float* out = (float*)d_out;

    const int IN_F = 4096;
    const int M    = in_sizes[0] / IN_F;       // 8192
    const int MPW  = 8;                        // m-rows per workgroup
    const int grid = (M + MPW - 1) / MPW;      // 1024 blocks

    kron_f32_wmma_kernel<<<grid, 128, 0, stream>>>(x, A, B, bias, out, MPW, M);
}